// GraphConvolution_25812753449812
// MI455X (gfx1250) — compile-verified
//
#include <hip/hip_runtime.h>
#include <stdint.h>

typedef float v2f __attribute__((ext_vector_type(2)));
typedef float v8f __attribute__((ext_vector_type(8)));

#define F_IN   512
#define F_OUT  512
#define LDS_STRIDE 516   // 512 + 4 pad -> bank-conflict-free column reads

// ---------------------------------------------------------------------------
// GEMM: support = X @ W using V_WMMA_F32_16X16X4_F32 (fp32 matrix path).
// Block = 256 threads (8 wave32 waves). Block tile: 16 rows x 256 cols.
// Each wave owns two 16x16 output tiles (reuses the A fragment twice).
// A strip (16 x 512 f32) staged in LDS via GLOBAL_LOAD_ASYNC_TO_LDS_B128
// (CDNA5 async memory->LDS path, tracked by ASYNCcnt).
// ---------------------------------------------------------------------------
__global__ __launch_bounds__(256) void gcn_gemm_wmma_f32(
    const float* __restrict__ X,
    const float* __restrict__ W,
    float* __restrict__ support)
{
    __shared__ float Atile[16 * LDS_STRIDE];

    const int row0    = blockIdx.y * 16;        // node-row tile (N % 16 == 0)
    const int colBase = blockIdx.x * 256;       // feature-column base
    const int tid     = threadIdx.x;

    // --- Stage A strip asynchronously: 16 rows x 512 f32 = 2048 x 16B chunks,
    //     8 chunks per thread, direct global->LDS (no VGPR bounce).
    {
        const uint32_t ldsBase = (uint32_t)(uintptr_t)&Atile[0];
        for (int i = tid; i < 16 * (F_IN / 4); i += 256) {
            const int r  = i / (F_IN / 4);
            const int c4 = i % (F_IN / 4);
            const uint64_t src =
                (uint64_t)(uintptr_t)(X + (size_t)(row0 + r) * F_IN + c4 * 4);
            const uint32_t dst =
                ldsBase + (uint32_t)((r * LDS_STRIDE + c4 * 4) * sizeof(float));
            asm volatile("global_load_async_to_lds_b128 %0, %1, off"
                         :: "v"(dst), "v"(src) : "memory");
        }
        asm volatile("s_wait_asynccnt 0x0" ::: "memory");
    }
    __syncthreads();

    const int wave = tid >> 5;
    const int lane = tid & 31;
    const int m    = lane & 15;           // A row within tile
    const int koff = (lane >> 4) * 2;     // lanes 16-31 hold K+2,K+3
    const int n0   = colBase + wave * 32 + (lane & 15);  // 1st col tile
    const int n1   = n0 + 16;                            // 2nd col tile

    v8f acc0 = {};
    v8f acc1 = {};
    const float* Arow = &Atile[m * LDS_STRIDE];

    for (int k0 = 0; k0 < F_IN; k0 += 4) {
        const int kA = k0 + koff;
        // A fragment (16x4): VGPR j = A[m][kA + j]
        v2f a;
        a.x = Arow[kA];
        a.y = Arow[kA + 1];
        // B fragments (4x16): VGPR j = W[kA + j][n]
        v2f b0, b1;
        b0.x = W[(size_t)(kA    ) * F_OUT + n0];
        b0.y = W[(size_t)(kA + 1) * F_OUT + n0];
        b1.x = W[(size_t)(kA    ) * F_OUT + n1];
        b1.y = W[(size_t)(kA + 1) * F_OUT + n1];

        acc0 = __builtin_amdgcn_wmma_f32_16x16x4_f32(
            false, a, false, b0, (short)0, acc0, false, false);
        acc1 = __builtin_amdgcn_wmma_f32_16x16x4_f32(
            false, a, false, b1, (short)0, acc1, false, false);
    }

    // C/D layout: VGPR r -> row (r + 8*(lane>>4)), col (lane&15)
    const int rbase = row0 + (lane >> 4) * 8;
    for (int r = 0; r < 8; ++r) {
        support[(size_t)(rbase + r) * F_OUT + n0] = acc0[r];
        support[(size_t)(rbase + r) * F_OUT + n1] = acc1[r];
    }
}

// ---------------------------------------------------------------------------
// out[n][f] = bias[f]  (reference adds bias to the segment sum)
// ---------------------------------------------------------------------------
__global__ void gcn_init_out(const float* __restrict__ bias,
                             float* __restrict__ out,
                             int total)
{
    const int idx = blockIdx.x * blockDim.x + threadIdx.x;
    if (idx < total)
        out[idx] = bias[idx & (F_OUT - 1)];
}

// ---------------------------------------------------------------------------
// SpMM scatter: for edge e, out[row_e] += w_e * support[col_e].
// 128 lanes per edge, 4 floats each (float4 gather, 4x atomic f32 add).
// Gather is ~6.5 GB; support (205 MB) nearly fits the 192 MB L2.
// ---------------------------------------------------------------------------
__global__ __launch_bounds__(256) void gcn_spmm_scatter(
    const int* __restrict__ rows,
    const int* __restrict__ cols,
    const float* __restrict__ ew,
    const float* __restrict__ support,
    float* __restrict__ out,
    int E)
{
    const int t = threadIdx.x & 127;               // feature chunk id
    const int e = blockIdx.x * 2 + (threadIdx.x >> 7);
    if (e >= E) return;

    const int   r  = rows[e];
    const int   c  = cols[e];
    const float we = ew[e];

    const float4 s =
        reinterpret_cast<const float4*>(support + (size_t)c * F_OUT)[t];
    float* o = out + (size_t)r * F_OUT + t * 4;

    atomicAdd(o + 0, we * s.x);
    atomicAdd(o + 1, we * s.y);
    atomicAdd(o + 2, we * s.z);
    atomicAdd(o + 3, we * s.w);
}

// ---------------------------------------------------------------------------
extern "C" void kernel_launch(void* const* d_in, const int* in_sizes, int n_in,
                              void* d_out, int out_size, void* d_ws, size_t ws_size,
                              hipStream_t stream)
{
    const float* x      = (const float*)d_in[0];   // [N, 512]
    const int*   eidx   = (const int*)  d_in[1];   // [2, E] flat
    const float* ew     = (const float*)d_in[2];   // [E]
    const float* weight = (const float*)d_in[3];   // [512, 512]
    const float* bias   = (const float*)d_in[4];   // [512]
    float*       out    = (float*)d_out;           // [N, 512]
    float*       support = (float*)d_ws;           // scratch: N*512 f32

    const int N = in_sizes[0] / F_IN;
    const int E = in_sizes[2];

    // 1) support = X @ W via fp32 WMMA
    dim3 gGemm(F_OUT / 256, N / 16);
    gcn_gemm_wmma_f32<<<gGemm, 256, 0, stream>>>(x, weight, support);

    // 2) out = broadcast(bias)
    const int total = N * F_OUT;
    gcn_init_out<<<(total + 255) / 256, 256, 0, stream>>>(bias, out, total);

    // 3) out[row] += w_e * support[col]
    gcn_spmm_scatter<<<(E + 1) / 2, 256, 0, stream>>>(
        eidx, eidx + E, ew, support, out, E);
}